// TangentPlane_41644002902481
// MI455X (gfx1250) — compile-verified
//
#include <hip/hip_runtime.h>
#include <cstdint>

// Problem constants (from the reference)
#define B_     16
#define C_     3
#define H_     1024
#define W_     2048
#define NPATCH 80
#define PATCH_ 16
#define PIX    (PATCH_ * PATCH_)   // 256 pixels per patch

// One workgroup per (patch, batch). 256 threads = 8 wave32 = one pixel each.
__global__ __launch_bounds__(PIX)
void tangent_bilinear_gather(const float* __restrict__ x,
                             const float* __restrict__ coords,
                             float* __restrict__ out)
{
    const int tid   = threadIdx.x;   // pixel index within patch: py*16+px
    const int patch = blockIdx.x;    // 0..79
    const int b     = blockIdx.y;    // 0..15

    // 2 KB staging buffer for this patch's (u,v) table.
    __shared__ __align__(16) float s_uv[PIX * 2];

    // --- CDNA5 async copy: global -> LDS, tracked by ASYNCcnt ---
    {
        const float* gsrc = coords + ((size_t)patch * PIX + (size_t)tid) * 2;
        // Low 32 bits of a generic pointer to LDS == the LDS byte address.
        unsigned lds_addr = (unsigned)(uintptr_t)(&s_uv[0]) + (unsigned)tid * 8u;
        asm volatile("global_load_async_to_lds_b64 %0, %1, off"
                     :: "v"(lds_addr), "v"(gsrc)
                     : "memory");
        asm volatile("s_wait_asynccnt 0" ::: "memory");
    }
    __syncthreads();

    const float u = s_uv[2 * tid + 0];
    const float v = s_uv[2 * tid + 1];

    const float x0f = floorf(u);
    const float y0f = floorf(v);
    const float wu  = u - x0f;
    const float wv  = v - y0f;

    // W is a power of two: mod == mask (u >= 0 in this problem anyway).
    int x0i = ((int)x0f) & (W_ - 1);
    int x1i = (x0i + 1) & (W_ - 1);
    int y0i = (int)y0f;
    y0i = y0i < 0 ? 0 : (y0i > (H_ - 1) ? (H_ - 1) : y0i);
    int y1i = (y0i + 1 > (H_ - 1)) ? (H_ - 1) : (y0i + 1);

    const int row0 = y0i * W_;
    const int row1 = y1i * W_;

    const size_t planeStride = (size_t)H_ * (size_t)W_;
    const float* plane0 = x + (size_t)(b * C_) * planeStride;

    // out[b, patch, c*256 + tid]
    float* o = out + ((size_t)b * NPATCH + (size_t)patch) * (size_t)(C_ * PIX) + tid;

#pragma unroll
    for (int c = 0; c < C_; ++c) {
        const float* p = plane0 + (size_t)c * planeStride;
        const float g00 = p[row0 + x0i];
        const float g01 = p[row0 + x1i];
        const float g10 = p[row1 + x0i];
        const float g11 = p[row1 + x1i];
        const float top = fmaf(wu, g01 - g00, g00);
        const float bot = fmaf(wu, g11 - g10, g10);
        o[(size_t)c * PIX] = fmaf(wv, bot - top, top);
    }
}

extern "C" void kernel_launch(void* const* d_in, const int* in_sizes, int n_in,
                              void* d_out, int out_size, void* d_ws, size_t ws_size,
                              hipStream_t stream) {
    (void)in_sizes; (void)n_in; (void)out_size; (void)d_ws; (void)ws_size;
    const float* x      = (const float*)d_in[0];
    const float* coords = (const float*)d_in[1];
    float*       out    = (float*)d_out;

    dim3 grid(NPATCH, B_, 1);
    dim3 block(PIX, 1, 1);
    tangent_bilinear_gather<<<grid, block, 0, stream>>>(x, coords, out);
}